// VimBlock_53755810677229
// MI455X (gfx1250) — compile-verified
//
#include <hip/hip_runtime.h>

// ---------------- dimensions (fixed by reference) ----------------
#define DIMX     512
#define D_INNER  1024
#define D_STATE  16
#define D_CONV   4
#define DT_RANK  32
#define BB       4
#define LL       1024
#define MROWS    (BB * LL)          // 4096 token rows

typedef float v2f __attribute__((ext_vector_type(2)));
typedef float v8f __attribute__((ext_vector_type(8)));

__device__ __forceinline__ float sigmoidf_(float x) {
    return 1.0f / (1.0f + __expf(-x));
}

// ---------------- LayerNorm + residual seed ----------------------
__global__ __launch_bounds__(256)
void ln_kernel(const float* __restrict__ x, const float* __restrict__ g,
               const float* __restrict__ be, float* __restrict__ xn,
               float* __restrict__ out) {
    int row = blockIdx.x;
    int tid = threadIdx.x;
    const float* xr = x + (size_t)row * DIMX;
    __shared__ float red[256];

    float v0 = xr[tid], v1 = xr[tid + 256];
    red[tid] = v0 + v1;
    __syncthreads();
    for (int o = 128; o > 0; o >>= 1) {
        if (tid < o) red[tid] += red[tid + o];
        __syncthreads();
    }
    float mu = red[0] * (1.0f / DIMX);
    __syncthreads();
    float d0 = v0 - mu, d1 = v1 - mu;
    red[tid] = d0 * d0 + d1 * d1;
    __syncthreads();
    for (int o = 128; o > 0; o >>= 1) {
        if (tid < o) red[tid] += red[tid + o];
        __syncthreads();
    }
    float rstd = rsqrtf(red[0] * (1.0f / DIMX) + 1e-5f);

    size_t base = (size_t)row * DIMX;
    xn[base + tid]       = d0 * rstd * g[tid]       + be[tid];
    xn[base + tid + 256] = d1 * rstd * g[tid + 256] + be[tid + 256];
    out[base + tid]       = v0;   // residual seed: out = x
    out[base + tid + 256] = v1;
}

// ---------------- fp32 WMMA GEMM: C(M,N) = A(M,K;lda) * W(N,K)^T --
// EPI: 0 = store, 1 = softplus(v + bias[n]) store, 2 = accumulate (+=)
// one wave -> 32(M) x 64(N) tile, K-step 4, software-pipelined loads
// (prefetch distance 1 so WMMA overlaps the next k-step's fragment loads)
template <int EPI>
__global__ __launch_bounds__(256)
void gemm_nt_wmma(const float* __restrict__ A, const float* __restrict__ W,
                  float* __restrict__ C, const float* __restrict__ bias,
                  int M, int N, int K, int lda) {
    int lane = threadIdx.x & 31;
    int wave = threadIdx.x >> 5;
    int task = blockIdx.x * 8 + wave;

    int nstrips = N >> 6;               // 64-wide N strips
    int mt = task / nstrips;            // 32-row M tile
    int ns = task % nstrips;
    if (mt * 32 >= M) return;           // wave-uniform

    int khalf = (lane >> 4) * 2;        // K sub-position: 0 or 2
    int nlo   = lane & 15;

    const float* a0p = A + (size_t)(mt * 32 + (lane & 15)) * lda + khalf;
    const float* a1p = a0p + (size_t)16 * lda;
    const float* w0p = W + (size_t)(ns * 64 +  0 + nlo) * K + khalf;
    const float* w1p = W + (size_t)(ns * 64 + 16 + nlo) * K + khalf;
    const float* w2p = W + (size_t)(ns * 64 + 32 + nlo) * K + khalf;
    const float* w3p = W + (size_t)(ns * 64 + 48 + nlo) * K + khalf;

    v8f acc[2][4];
    #pragma unroll
    for (int i = 0; i < 2; ++i)
        #pragma unroll
        for (int t = 0; t < 4; ++t)
            acc[i][t] = v8f{0.f,0.f,0.f,0.f,0.f,0.f,0.f,0.f};

    // prologue: fragments for k = 0
    v2f a0 = *(const v2f*)a0p, a1 = *(const v2f*)a1p;
    v2f b0 = *(const v2f*)w0p, b1 = *(const v2f*)w1p;
    v2f b2 = *(const v2f*)w2p, b3 = *(const v2f*)w3p;

    for (int k = 4; k < K; k += 4) {
        // issue next k-step's loads before this step's WMMAs
        v2f a0n = *(const v2f*)(a0p + k);
        v2f a1n = *(const v2f*)(a1p + k);
        v2f b0n = *(const v2f*)(w0p + k);
        v2f b1n = *(const v2f*)(w1p + k);
        v2f b2n = *(const v2f*)(w2p + k);
        v2f b3n = *(const v2f*)(w3p + k);

        acc[0][0] = __builtin_amdgcn_wmma_f32_16x16x4_f32(false, a0, false, b0, (short)0, acc[0][0], false, false);
        acc[1][0] = __builtin_amdgcn_wmma_f32_16x16x4_f32(false, a1, false, b0, (short)0, acc[1][0], false, false);
        acc[0][1] = __builtin_amdgcn_wmma_f32_16x16x4_f32(false, a0, false, b1, (short)0, acc[0][1], false, false);
        acc[1][1] = __builtin_amdgcn_wmma_f32_16x16x4_f32(false, a1, false, b1, (short)0, acc[1][1], false, false);
        acc[0][2] = __builtin_amdgcn_wmma_f32_16x16x4_f32(false, a0, false, b2, (short)0, acc[0][2], false, false);
        acc[1][2] = __builtin_amdgcn_wmma_f32_16x16x4_f32(false, a1, false, b2, (short)0, acc[1][2], false, false);
        acc[0][3] = __builtin_amdgcn_wmma_f32_16x16x4_f32(false, a0, false, b3, (short)0, acc[0][3], false, false);
        acc[1][3] = __builtin_amdgcn_wmma_f32_16x16x4_f32(false, a1, false, b3, (short)0, acc[1][3], false, false);

        a0 = a0n; a1 = a1n; b0 = b0n; b1 = b1n; b2 = b2n; b3 = b3n;
    }
    // epilogue k-step
    acc[0][0] = __builtin_amdgcn_wmma_f32_16x16x4_f32(false, a0, false, b0, (short)0, acc[0][0], false, false);
    acc[1][0] = __builtin_amdgcn_wmma_f32_16x16x4_f32(false, a1, false, b0, (short)0, acc[1][0], false, false);
    acc[0][1] = __builtin_amdgcn_wmma_f32_16x16x4_f32(false, a0, false, b1, (short)0, acc[0][1], false, false);
    acc[1][1] = __builtin_amdgcn_wmma_f32_16x16x4_f32(false, a1, false, b1, (short)0, acc[1][1], false, false);
    acc[0][2] = __builtin_amdgcn_wmma_f32_16x16x4_f32(false, a0, false, b2, (short)0, acc[0][2], false, false);
    acc[1][2] = __builtin_amdgcn_wmma_f32_16x16x4_f32(false, a1, false, b2, (short)0, acc[1][2], false, false);
    acc[0][3] = __builtin_amdgcn_wmma_f32_16x16x4_f32(false, a0, false, b3, (short)0, acc[0][3], false, false);
    acc[1][3] = __builtin_amdgcn_wmma_f32_16x16x4_f32(false, a1, false, b3, (short)0, acc[1][3], false, false);

    // C/D layout: vgpr j -> M = base + j + 8*(lane>=16), N = lane%16
    #pragma unroll
    for (int i = 0; i < 2; ++i) {
        int mbase = mt * 32 + i * 16 + (lane >> 4) * 8;
        #pragma unroll
        for (int t = 0; t < 4; ++t) {
            int ncol = ns * 64 + t * 16 + nlo;
            #pragma unroll
            for (int j = 0; j < 8; ++j) {
                size_t off = (size_t)(mbase + j) * N + ncol;
                float v = acc[i][t][j];
                if (EPI == 1) {
                    v += bias[ncol];
                    v = (v > 20.0f) ? v : log1pf(__expf(v));   // softplus
                }
                if (EPI == 2) C[off] += v;
                else          C[off] = v;
            }
        }
    }
}

// ---------------- depthwise causal conv (4 taps) + SiLU ----------
__global__ __launch_bounds__(256)
void conv_silu_kernel(const float* __restrict__ xz, const float* __restrict__ w,
                      const float* __restrict__ b, float* __restrict__ xc,
                      int forward) {
    int idx = blockIdx.x * blockDim.x + threadIdx.x;   // [0, MROWS*D_INNER)
    int d = idx & (D_INNER - 1);
    int m = idx >> 10;
    int bi = m >> 10;
    int l  = m & (LL - 1);

    float acc = b[d];
    #pragma unroll
    for (int j = 0; j < D_CONV; ++j) {
        int src = forward ? (l - (D_CONV - 1) + j) : (l + (D_CONV - 1) - j);
        if (src >= 0 && src < LL) {
            size_t sm = (size_t)(bi * LL + src) * (2 * D_INNER) + d;
            acc += w[d * D_CONV + j] * xz[sm];
        }
    }
    xc[(size_t)m * D_INNER + d] = acc * sigmoidf_(acc);
}

// ---------------- selective scan, fused epilogue ------------------
// one block = one batch b + 256-wide d chunk (b, d-base from blockIdx only
// so per-step B/C addresses are provably uniform -> SMEM broadcast loads).
__global__ __launch_bounds__(256)
void scan_kernel(const float* __restrict__ dt, const float* __restrict__ xc,
                 const float* __restrict__ xz, const float* __restrict__ dbl,
                 const float* __restrict__ A_log, const float* __restrict__ Dp,
                 float* __restrict__ ys, int forward) {
    int bi = blockIdx.x >> 2;                               // uniform batch
    int d  = ((blockIdx.x & 3) << 8) + threadIdx.x;         // d chunk

    float a[D_STATE];
    #pragma unroll
    for (int n = 0; n < D_STATE; ++n)
        a[n] = -__expf(A_log[d * D_STATE + n]);
    float Dd = Dp[d];

    float h[D_STATE];
    #pragma unroll
    for (int n = 0; n < D_STATE; ++n) h[n] = 0.0f;

    for (int s = 0; s < LL; ++s) {
        int l = forward ? s : (LL - 1 - s);
        size_t m = (size_t)bi * LL + l;                     // block-uniform
        float dtv = dt[m * D_INNER + d];
        float xcv = xc[m * D_INNER + d];
        float zv  = xz[m * (2 * D_INNER) + D_INNER + d];
        const float* bc = dbl + m * (DT_RANK + 2 * D_STATE);  // uniform addr

        if (s + 8 < LL) {                                   // stream prefetch
            int lp = forward ? (s + 8) : (LL - 1 - (s + 8));
            size_t mp = (size_t)bi * LL + lp;
            __builtin_prefetch(dt + mp * D_INNER + d, 0, 1);
            __builtin_prefetch(xc + mp * D_INNER + d, 0, 1);
            __builtin_prefetch(xz + mp * (2 * D_INNER) + D_INNER + d, 0, 1);
        }

        float dbu = dtv * xcv;
        float y = 0.0f;
        #pragma unroll
        for (int n = 0; n < D_STATE; ++n) {
            float dA = __expf(dtv * a[n]);
            h[n] = dA * h[n] + dbu * bc[DT_RANK + n];          // B
            y += h[n] * bc[DT_RANK + D_STATE + n];             // C
        }
        y = (y + Dd * xcv) * (zv * sigmoidf_(zv));
        ys[m * D_INNER + d] = y;
    }
}

// ---------------- launcher ---------------------------------------
extern "C" void kernel_launch(void* const* d_in, const int* in_sizes, int n_in,
                              void* d_out, int out_size, void* d_ws, size_t ws_size,
                              hipStream_t stream) {
    (void)in_sizes; (void)n_in; (void)out_size; (void)ws_size;
    const float* x  = (const float*)d_in[0];
    const float* g  = (const float*)d_in[1];
    const float* be = (const float*)d_in[2];
    float* out = (float*)d_out;

    // workspace layout (floats), ~93 MB total
    float* ws  = (float*)d_ws;
    float* xn  = ws;                              // 4096*512
    float* xz  = xn  + (size_t)MROWS * DIMX;      // 4096*2048
    float* xc  = xz  + (size_t)MROWS * 2 * D_INNER;
    float* dbl = xc  + (size_t)MROWS * D_INNER;   // 4096*64
    float* dtb = dbl + (size_t)MROWS * (DT_RANK + 2 * D_STATE);
    float* ysb = dtb + (size_t)MROWS * D_INNER;

    ln_kernel<<<MROWS, 256, 0, stream>>>(x, g, be, xn, out);

    for (int dir = 0; dir < 2; ++dir) {
        void* const* P = d_in + 3 + dir * 9;   // params dict leaves, decl order
        const float* in_proj  = (const float*)P[0];   // (2048, 512)
        const float* conv_w   = (const float*)P[1];   // (1024, 4)
        const float* conv_b   = (const float*)P[2];   // (1024,)
        const float* x_proj   = (const float*)P[3];   // (64, 1024)
        const float* dt_w     = (const float*)P[4];   // (1024, 32)
        const float* dt_b     = (const float*)P[5];   // (1024,)
        const float* A_log    = (const float*)P[6];   // (1024, 16)
        const float* Dp       = (const float*)P[7];   // (1024,)
        const float* out_proj = (const float*)P[8];   // (512, 1024)
        int fwd = (dir == 0);

        // xz(4096,2048) = xn * in_proj^T : (4096/32)*(2048/64)=4096 waves
        gemm_nt_wmma<0><<<512, 256, 0, stream>>>(xn, in_proj, xz, nullptr,
                                                 MROWS, 2 * D_INNER, DIMX, DIMX);
        conv_silu_kernel<<<(MROWS * D_INNER) / 256, 256, 0, stream>>>(
            xz, conv_w, conv_b, xc, fwd);
        // dbl(4096,64) = xc * x_proj^T : 128 waves
        gemm_nt_wmma<0><<<16, 256, 0, stream>>>(xc, x_proj, dbl, nullptr,
                                                MROWS, DT_RANK + 2 * D_STATE,
                                                D_INNER, D_INNER);
        // dt(4096,1024) = softplus(dbl[:,:32] * dt_w^T + dt_b) : 2048 waves
        gemm_nt_wmma<1><<<256, 256, 0, stream>>>(dbl, dt_w, dtb, dt_b,
                                                 MROWS, D_INNER, DT_RANK,
                                                 DT_RANK + 2 * D_STATE);
        scan_kernel<<<(BB * D_INNER) / 256, 256, 0, stream>>>(
            dtb, xc, xz, dbl, A_log, Dp, ysb, fwd);
        // out(4096,512) += ys * out_proj^T : 1024 waves
        gemm_nt_wmma<2><<<128, 256, 0, stream>>>(ysb, out_proj, out, nullptr,
                                                 MROWS, DIMX, D_INNER, D_INNER);
    }
}